// Torch_compute_derivatives_with_lon_lat_18588618457628
// MI455X (gfx1250) — compile-verified
//
#include <hip/hip_runtime.h>
#include <math.h>
#include <stdint.h>

// Problem constants (match reference setup_inputs)
#define BATCH 4
#define DT    7
#define NPLANE (BATCH*DT)     // 28 planes
#define H 512
#define W 512

// Tiling: 32x32 output tile, symmetric halo 8 (needed: 1 grad + 5 heat + 1 grad = 7)
#define T     32
#define RAD   8
#define SPAN  (T + 2*RAD)     // 48 rows and cols held in LDS
#define NT    256             // threads per block (8 wave32)
#define NCHUNK (SPAN/4)       // 12 B128 chunks per 48-float row

#define R_EARTH 6371000.0f
#define OMEGA   7.2921e-05f
#define ALPHA   9.81f
#define EPS     1e-10f

// single-step reflect (pad=1 reflect): -1 -> 1, n -> n-2
__device__ __forceinline__ int refl(int i, int n) {
    return (i < 0) ? -i : ((i >= n) ? (2*n - 2 - i) : i);
}

// CDNA5 async global->LDS copies (ASYNCcnt-tracked). GVS addressing:
// LDS address from vdst ; global = saddr(64b SGPR pair) + vaddr(32b offset)
__device__ __forceinline__ void async_ld_b32(uint32_t lds_byte_off,
                                             const float* base,
                                             uint32_t byte_off) {
    asm volatile("global_load_async_to_lds_b32 %0, %1, %2"
                 :: "v"(lds_byte_off), "v"(byte_off), "s"(base)
                 : "memory");
}
__device__ __forceinline__ void async_ld_b128(uint32_t lds_byte_off,
                                              const float* base,
                                              uint32_t byte_off) {
    asm volatile("global_load_async_to_lds_b128 %0, %1, %2"
                 :: "v"(lds_byte_off), "v"(byte_off), "s"(base)
                 : "memory");
}
__device__ __forceinline__ void wait_async_all() {
    asm volatile("s_wait_asynccnt 0" ::: "memory");
}

__global__ __launch_bounds__(NT)
void derive_fused_kernel(const float* __restrict__ ssh,
                         const float* __restrict__ lat,
                         const float* __restrict__ lon,
                         const float* __restrict__ mask,
                         float* __restrict__ out) {
    __shared__ __align__(16) float s_in [SPAN][SPAN];
    __shared__ __align__(16) float s_msk[SPAN][SPAN];
    __shared__ __align__(16) float s_u0 [SPAN][SPAN];
    __shared__ __align__(16) float s_u1 [SPAN][SPAN];
    __shared__ __align__(16) float s_v0 [SPAN][SPAN];
    __shared__ __align__(16) float s_v1 [SPAN][SPAN];
    __shared__ float s_gu [SPAN];   // -ALPHA / (f_c * dy)
    __shared__ float s_gv [SPAN];   //  ALPHA / (f_c * dx)
    __shared__ float s_idx[SPAN];   // 1/dx per row
    __shared__ float s_idy[SPAN];   // 1/dy per row

    const int tid = threadIdx.x;
    const int p  = blockIdx.z;        // plane index = b*DT + t
    const int b  = p / DT;
    const int y0 = blockIdx.y * T;
    const int x0 = blockIdx.x * T;
    const int oy = y0 - RAD;
    const int ox = x0 - RAD;

    const float* sshp = ssh  + (size_t)p * (H * W);
    const float* mskp = mask + (size_t)p * (H * W);

    // ---- Stage 0: async DMA of ssh + mask 48x48 tiles into LDS
    const bool interior = (oy >= 0) && (oy + SPAN <= H) && (ox >= 0) && (ox + SPAN <= W);
    if (interior) {
        // fast path: full rows, 16B-aligned ((x0-8)*4 is 32B aligned), B128 DMA
        for (int i = tid; i < SPAN * NCHUNK; i += NT) {
            int l = i / NCHUNK, ch = i - l * NCHUNK;
            uint32_t goff = (uint32_t)((((oy + l) * W) + ox + ch * 4) * 4);
            async_ld_b128((uint32_t)(uintptr_t)&s_in [l][ch * 4], sshp, goff);
            async_ld_b128((uint32_t)(uintptr_t)&s_msk[l][ch * 4], mskp, goff);
        }
    } else {
        // border path: per-element with bounds checks (reflect handled at read time)
        for (int i = tid; i < SPAN * SPAN; i += NT) {
            int l = i / SPAN, c = i - l * SPAN;
            int gy = oy + l, gx = ox + c;
            if ((unsigned)gy < (unsigned)H && (unsigned)gx < (unsigned)W) {
                uint32_t goff = (uint32_t)((gy * W + gx) * 4);
                async_ld_b32((uint32_t)(uintptr_t)&s_in [l][c], sshp, goff);
                async_ld_b32((uint32_t)(uintptr_t)&s_msk[l][c], mskp, goff);
            }
        }
    }

    // ---- Per-row geometry (overlaps the async DMA)
    if (tid < SPAN) {
        float dlat = lat[1] - lat[0];
        float dlon = lon[1] - lon[0];
        int gy = oy + tid;
        int cy = gy < 0 ? 0 : (gy > H - 1 ? H - 1 : gy);
        float la = lat[b * H + cy];
        // dx from dlon: a = (cos(lat)*sin(dlon/2))^2
        float s1 = cosf(la) * sinf(0.5f * dlon);
        float a1 = s1 * s1;
        float dx = 2.0f * R_EARTH * atan2f(sqrtf(a1 + EPS), sqrtf(1.0f - a1 + EPS));
        // dy from dlat: a = sin(dlat/2)^2
        float s2 = sinf(0.5f * dlat);
        float a2 = s2 * s2;
        float dy = 2.0f * R_EARTH * atan2f(sqrtf(a2 + EPS), sqrtf(1.0f - a2 + EPS));
        float fc = 2.0f * OMEGA * sinf(la);
        s_idx[tid] = 1.0f / dx;
        s_idy[tid] = 1.0f / dy;
        s_gu[tid]  = -ALPHA / (fc * dy);
        s_gv[tid]  =  ALPHA / (fc * dx);
    }

    wait_async_all();
    __syncthreads();

    // ---- Stage 1: geostrophic velocities over halo-6 region (44x44)
    {
        const int lo = RAD - 6, sp = T + 12;   // lo=2, sp=44
        for (int i = tid; i < sp * sp; i += NT) {
            int r = i / sp;
            int l = r + lo, c = i - r * sp + lo;
            int gy = oy + l, gx = ox + c;
            if ((unsigned)gy < (unsigned)H && (unsigned)gx < (unsigned)W) {
                int lm = refl(gy - 1, H) - oy, lp = refl(gy + 1, H) - oy;
                int cm = refl(gx - 1, W) - ox, cp = refl(gx + 1, W) - ox;
                float ctr = s_in[l][c];
                float gyv = -0.7f * s_in[lm][c] + 0.4f * ctr + 0.3f * s_in[lp][c];
                float gxv = -0.7f * s_in[l][cm] + 0.4f * ctr + 0.3f * s_in[l][cp];
                s_u0[l][c] = s_gu[l] * gyv;     // u_geo
                s_v0[l][c] = s_gv[l] * gxv;     // v_geo
            }
        }
    }
    __syncthreads();

    // ---- Stage 2: five masked heat-diffusion iterations (ping-pong in LDS)
    float (*uin)[SPAN]  = s_u0;
    float (*uout)[SPAN] = s_u1;
    float (*vin)[SPAN]  = s_v0;
    float (*vout)[SPAN] = s_v1;
    #pragma unroll
    for (int it = 0; it < 5; ++it) {
        int k  = 5 - it;          // output halo of this iteration
        int lo = RAD - k;         // 3 + it
        int sp = T + 2 * k;       // 42 - 2*it
        for (int i = tid; i < sp * sp; i += NT) {
            int r = i / sp;
            int l = r + lo, c = i - r * sp + lo;
            int gy = oy + l, gx = ox + c;
            if ((unsigned)gy < (unsigned)H && (unsigned)gx < (unsigned)W) {
                int lm = refl(gy - 1, H) - oy, lp = refl(gy + 1, H) - oy;
                int cm = refl(gx - 1, W) - ox, cp = refl(gx + 1, W) - ox;
                float m = 0.05f * s_msk[l][c];   // 0.2 * 0.25 * mask
                float cu = uin[lm][c] + uin[lp][c] + uin[l][cm] + uin[l][cp];
                uout[l][c] = 0.8f * uin[l][c] + m * cu;
                float cv = vin[lm][c] + vin[lp][c] + vin[l][cm] + vin[l][cp];
                vout[l][c] = 0.8f * vin[l][c] + m * cv;
            }
        }
        __syncthreads();
        float (*tu)[SPAN] = uin; uin = uout; uout = tu;
        float (*tv)[SPAN] = vin; vin = vout; vout = tv;
    }
    // final smoothed fields are in uin / vin, valid over halo-1 region

    // ---- Stage 3: derivatives, div/curl/strain; non-temporal channel stores
    const size_t CH = (size_t)NPLANE * H * W;   // per-channel stride
    for (int i = tid; i < T * T; i += NT) {
        int r = i / T;
        int l = r + RAD, c = i - r * T + RAD;
        int gy = oy + l, gx = ox + c;
        if ((unsigned)gy < (unsigned)H && (unsigned)gx < (unsigned)W) {
            int lm = refl(gy - 1, H) - oy, lp = refl(gy + 1, H) - oy;
            int cm = refl(gx - 1, W) - ox, cp = refl(gx + 1, W) - ox;
            float uu = uin[l][c], vv = vin[l][c];
            float idx = s_idx[l], idy = s_idy[l];
            float du_dx = idx * (-0.7f * uin[l][cm] + 0.4f * uu + 0.3f * uin[l][cp]);
            float du_dy = idy * (-0.7f * uin[lm][c] + 0.4f * uu + 0.3f * uin[lp][c]);
            float dv_dx = idx * (-0.7f * vin[l][cm] + 0.4f * vv + 0.3f * vin[l][cp]);
            float dv_dy = idy * (-0.7f * vin[lm][c] + 0.4f * vv + 0.3f * vin[lp][c]);
            float dv    = du_dx + dv_dy;
            float curl  = du_dy - dv_dx;
            float sh    = dv_dx + du_dy;
            float nr    = du_dx - dv_dy;
            float strain = sqrtf(sh * sh + nr * nr + EPS);
            size_t off = (size_t)p * (H * W) + (size_t)gy * W + gx;
            __builtin_nontemporal_store(uu,     &out[off]);
            __builtin_nontemporal_store(vv,     &out[CH + off]);
            __builtin_nontemporal_store(dv,     &out[2 * CH + off]);
            __builtin_nontemporal_store(curl,   &out[3 * CH + off]);
            __builtin_nontemporal_store(strain, &out[4 * CH + off]);
        }
    }
}

extern "C" void kernel_launch(void* const* d_in, const int* in_sizes, int n_in,
                              void* d_out, int out_size, void* d_ws, size_t ws_size,
                              hipStream_t stream) {
    (void)in_sizes; (void)n_in; (void)out_size; (void)d_ws; (void)ws_size;
    const float* ssh  = (const float*)d_in[0];
    const float* lat  = (const float*)d_in[1];
    const float* lon  = (const float*)d_in[2];
    const float* mask = (const float*)d_in[3];
    dim3 grid(W / T, H / T, NPLANE);   // 16 x 16 x 28 tiles
    derive_fused_kernel<<<grid, NT, 0, stream>>>(ssh, lat, lon, mask, (float*)d_out);
}